// EGLRGCNModel_39779987096176
// MI455X (gfx1250) — compile-verified
//
#include <hip/hip_runtime.h>

typedef float v2f __attribute__((ext_vector_type(2)));
typedef float v8f __attribute__((ext_vector_type(8)));

#define N_NODES 50000
#define N_EDGES 800000
#define IN_DIM  64
#define HID_DIM 64
#define OUT_DIM 16
#define N_BASES 4

// ---------------- zero scratch / output ----------------
__global__ void __launch_bounds__(256) zero_kernel(float* __restrict__ p, int n) {
  int i = blockIdx.x * blockDim.x + threadIdx.x;
  int stride = gridDim.x * blockDim.x;
  for (; i < n; i += stride) p[i] = 0.0f;
}

// ---------------- Layer 1 GEMM: Y1[n, b*64+o] = sum_i X[n,i]*bases1[b,i,o] ----------------
// One wave32 per 16x16 output tile, V_WMMA_F32_16X16X4_F32, K=64 stepped by 4.
// Tiles: M = 50000/16 = 3125 (exact), N = 256/16 = 16 -> 50000 tiles, 8 waves/block.
__global__ void __launch_bounds__(256) gemm_xb1(const float* __restrict__ X,
                                                const float* __restrict__ B1,
                                                float* __restrict__ Y1) {
  const int wave  = threadIdx.x >> 5;
  const int lane  = threadIdx.x & 31;
  const int tile  = blockIdx.x * 8 + wave;          // 0..49999
  const int mtile = tile >> 4;
  const int ntile = tile & 15;
  const int m0    = mtile << 4;
  const int col   = (ntile << 4) + (lane & 15);     // 0..255
  const int khalf = (lane >> 4) << 1;               // lanes 0-15: K+{0,1}; 16-31: K+{2,3}

  const float* arow = X + (size_t)(m0 + (lane & 15)) * IN_DIM;
  // B(i, j) with j = b*64+o maps to bases1[b*4096 + i*64 + o]
  const float* bcol = B1 + (size_t)(col >> 6) * (IN_DIM * HID_DIM) + (col & 63);

  v8f acc = {0.f, 0.f, 0.f, 0.f, 0.f, 0.f, 0.f, 0.f};
#pragma unroll
  for (int k = 0; k < IN_DIM; k += 4) {
    v2f a = *(const v2f*)(arow + k + khalf);        // aligned b64 load
    v2f b;
    b.x = bcol[(size_t)(k + khalf)     * HID_DIM];
    b.y = bcol[(size_t)(k + khalf + 1) * HID_DIM];
    acc = __builtin_amdgcn_wmma_f32_16x16x4_f32(false, a, false, b,
                                                (short)0, acc, false, false);
  }
  const int rbase = m0 + ((lane >> 4) << 3);
#pragma unroll
  for (int v = 0; v < 8; ++v)
    Y1[(size_t)(rbase + v) * (N_BASES * HID_DIM) + col] = acc[v];
}

// ---------------- Layer 1 edge kernel ----------------
// tid>>6 = edge (wave-uniform), tid&63 = output channel.
// msg[o] = norm * sum_b coef[etype,b] * Y1[src, b, o]; atomic scatter-add by dst.
__global__ void __launch_bounds__(256) edge_pass1(const int* __restrict__ src,
                                                  const int* __restrict__ dst,
                                                  const int* __restrict__ etype,
                                                  const float* __restrict__ enorm,
                                                  const float* __restrict__ coef,
                                                  const float* __restrict__ Y1,
                                                  float* __restrict__ H1) {
  const int tid = blockIdx.x * 256 + threadIdx.x;
  const int e = tid >> 6;
  const int o = tid & 63;
  const int s = src[e];
  const int d = dst[e];
  const int t = etype[e];
  const float nv = enorm[e];
  const float* c = coef + t * N_BASES;
  const float* y = Y1 + (size_t)s * (N_BASES * HID_DIM) + o;
  float acc = c[0] * y[0] + c[1] * y[64] + c[2] * y[128] + c[3] * y[192];
  __hip_atomic_fetch_add(&H1[(size_t)d * HID_DIM + o], nv * acc,
                         __ATOMIC_RELAXED, __HIP_MEMORY_SCOPE_AGENT);
}

// ---------------- Layer 2 GEMM: Y2[n, b*16+o] = sum_i relu(H1[n,i]+bias1[i])*bases2[b,i,o] ----
// Tiles: M = 3125, N = 64/16 = 4 -> 12500 tiles.
__global__ void __launch_bounds__(256) gemm_hb2(const float* __restrict__ H1,
                                                const float* __restrict__ bias1,
                                                const float* __restrict__ B2,
                                                float* __restrict__ Y2) {
  const int wave  = threadIdx.x >> 5;
  const int lane  = threadIdx.x & 31;
  const int tile  = blockIdx.x * 8 + wave;
  if (tile >= 3125 * 4) return;
  const int mtile = tile >> 2;
  const int ntile = tile & 3;
  const int m0    = mtile << 4;
  const int col   = (ntile << 4) + (lane & 15);     // 0..63
  const int khalf = (lane >> 4) << 1;

  const float* arow = H1 + (size_t)(m0 + (lane & 15)) * HID_DIM;
  // B2(i, j) with j = b*16+o maps to bases2[b*1024 + i*16 + o]
  const float* bcol = B2 + (size_t)(col >> 4) * (HID_DIM * OUT_DIM) + (col & 15);

  v8f acc = {0.f, 0.f, 0.f, 0.f, 0.f, 0.f, 0.f, 0.f};
#pragma unroll
  for (int k = 0; k < HID_DIM; k += 4) {
    const int kk = k + khalf;
    v2f a;
    a.x = fmaxf(arow[kk]     + bias1[kk],     0.0f);  // fused bias+relu epilogue of layer 1
    a.y = fmaxf(arow[kk + 1] + bias1[kk + 1], 0.0f);
    v2f b;
    b.x = bcol[(size_t)kk       * OUT_DIM];
    b.y = bcol[(size_t)(kk + 1) * OUT_DIM];
    acc = __builtin_amdgcn_wmma_f32_16x16x4_f32(false, a, false, b,
                                                (short)0, acc, false, false);
  }
  const int rbase = m0 + ((lane >> 4) << 3);
#pragma unroll
  for (int v = 0; v < 8; ++v)
    Y2[(size_t)(rbase + v) * (N_BASES * OUT_DIM) + col] = acc[v];
}

// ---------------- Layer 2 edge kernel ----------------
__global__ void __launch_bounds__(256) edge_pass2(const int* __restrict__ src,
                                                  const int* __restrict__ dst,
                                                  const int* __restrict__ etype,
                                                  const float* __restrict__ enorm,
                                                  const float* __restrict__ coef,
                                                  const float* __restrict__ Y2,
                                                  float* __restrict__ out) {
  const int tid = blockIdx.x * 256 + threadIdx.x;
  const int e = tid >> 4;
  const int o = tid & 15;
  const int s = src[e];
  const int d = dst[e];
  const int t = etype[e];
  const float nv = enorm[e];
  const float* c = coef + t * N_BASES;
  const float* y = Y2 + (size_t)s * (N_BASES * OUT_DIM) + o;
  float acc = c[0] * y[0] + c[1] * y[16] + c[2] * y[32] + c[3] * y[48];
  __hip_atomic_fetch_add(&out[(size_t)d * OUT_DIM + o], nv * acc,
                         __ATOMIC_RELAXED, __HIP_MEMORY_SCOPE_AGENT);
}

// ---------------- final bias + relu (in place on d_out) ----------------
__global__ void __launch_bounds__(256) bias_relu_out(float* __restrict__ out,
                                                     const float* __restrict__ bias2) {
  const int i = blockIdx.x * 256 + threadIdx.x;   // 800000 elems exactly
  out[i] = fmaxf(out[i] + bias2[i & 15], 0.0f);
}

extern "C" void kernel_launch(void* const* d_in, const int* in_sizes, int n_in,
                              void* d_out, int out_size, void* d_ws, size_t ws_size,
                              hipStream_t stream) {
  const float* feats  = (const float*)d_in[0];
  const int*   src    = (const int*)  d_in[1];
  const int*   dst    = (const int*)  d_in[2];
  const int*   etype  = (const int*)  d_in[3];
  const float* enorm  = (const float*)d_in[4];
  const float* bases1 = (const float*)d_in[5];
  const float* coef1  = (const float*)d_in[6];
  const float* bias1  = (const float*)d_in[7];
  const float* bases2 = (const float*)d_in[8];
  const float* coef2  = (const float*)d_in[9];
  const float* bias2  = (const float*)d_in[10];
  float* out = (float*)d_out;

  char* ws = (char*)d_ws;
  float* Y1 = (float*)ws;                                          // 50000*256 f32 = 51.2 MB
  float* H1 = (float*)(ws + (size_t)N_NODES * 256 * sizeof(float)); // 50000*64  f32 = 12.8 MB
  float* Y2 = (float*)(ws + (size_t)N_NODES * 320 * sizeof(float)); // 50000*64  f32 = 12.8 MB

  // zero accumulators (atomics accumulate; must re-init every call)
  zero_kernel<<<2048, 256, 0, stream>>>(H1, N_NODES * HID_DIM);
  zero_kernel<<<2048, 256, 0, stream>>>(out, N_NODES * OUT_DIM);

  // layer 1: per-node basis transform (WMMA), then edge combine/scatter
  gemm_xb1<<<50000 / 8, 256, 0, stream>>>(feats, bases1, Y1);
  edge_pass1<<<(N_EDGES * 64) / 256, 256, 0, stream>>>(src, dst, etype, enorm, coef1, Y1, H1);

  // layer 2: fused relu(h+bias1) into WMMA A-load, then edge combine/scatter
  gemm_hb2<<<(12500 + 7) / 8, 256, 0, stream>>>(H1, bias1, bases2, Y2);
  edge_pass2<<<(N_EDGES * 16) / 256, 256, 0, stream>>>(src, dst, etype, enorm, coef2, Y2, out);

  // final epilogue
  bias_relu_out<<<(N_NODES * OUT_DIM) / 256, 256, 0, stream>>>(out, bias2);
}